// MinibatchDiscrimination_48043504173685
// MI455X (gfx1250) — compile-verified
//
#include <hip/hip_runtime.h>
#include <hip/hip_bf16.h>

// MinibatchDiscrimination for MI455X (gfx1250, wave32).
// M = x @ T with V_WMMA_F32_16X16X4_F32 (exact f32, matches f32 reference).
// Register-tiled: each wave owns a 64x16 output tile (4 accumulators) so one
// B fragment feeds 4 WMMAs -> 1.5 vmem instr per wmma instead of 3.
// Pairwise L1 + exp stage is VALU/LDS (not expressible as matmul).

typedef __attribute__((ext_vector_type(2))) float v2f;
typedef __attribute__((ext_vector_type(8))) float v8f;

#define B_SZ    128
#define IN_F    1024
#define OUT_F   128
#define INTER_F 32
#define NCOL    (OUT_F * INTER_F)   // 4096
#define OUTW    (IN_F + OUT_F)      // 1152

// ---------------------------------------------------------------------------
// Kernel 1: M[128,4096] = x[128,1024] @ T[1024,4096], f32 WMMA 16x16x4.
// grid.x = 4096/64 = 64; block = 256 threads = 8 waves.
// Wave w: m-half = w&1 (rows 64*(w&1) .. +64), n-tile = w>>1 within block.
// Each wave: 4 M-tiles x 1 N-tile -> 4 v8f accumulators.
// A frag (16x4 f32, 2 VGPR): lanes 0-15 hold (row=m+l, K=k..k+1),
//                            lanes 16-31 hold (row=m+l, K=k+2..k+3).
// B frag (4x16 f32, 2 VGPR): lanes 0-15 hold (K=k..k+1,  col=n0+l),
//                            lanes 16-31 hold (K=k+2..k+3, col=n0+l).
// D (16x16 f32, 8 VGPR): vgpr r, lanes 0-15 -> (m+r, n0+l);
//                                lanes 16-31 -> (m+8+r, n0+l).
// ---------------------------------------------------------------------------
__global__ __launch_bounds__(256) void gemm_wmma_f32(
    const float* __restrict__ x, const float* __restrict__ T,
    float* __restrict__ M) {
  const int lane  = threadIdx.x & 31;
  const int wave  = threadIdx.x >> 5;      // 0..7
  const int l     = lane & 15;
  const int hi    = lane >> 4;             // 0: K pair (k,k+1); 1: (k+2,k+3)
  const int mhalf = wave & 1;              // rows [64*mhalf, 64*mhalf+64)
  const int nt    = wave >> 1;             // 0..3 N-tile within block

  const int m0 = mhalf * 64;
  const int n0 = blockIdx.x * 64 + nt * 16;

  const float* arow = x + (size_t)(m0 + l) * IN_F + 2 * hi;    // contiguous K
  const float* bcol = T + (size_t)(2 * hi) * NCOL + (n0 + l);  // strided K

  v8f acc0 = {}, acc1 = {}, acc2 = {}, acc3 = {};
#pragma unroll 2
  for (int k = 0; k < IN_F; k += 4) {
    v2f a0 = *(const v2f*)(arow + k);                 // rows m0..m0+15
    v2f a1 = *(const v2f*)(arow + 16 * IN_F + k);     // rows m0+16..
    v2f a2 = *(const v2f*)(arow + 32 * IN_F + k);     // rows m0+32..
    v2f a3 = *(const v2f*)(arow + 48 * IN_F + k);     // rows m0+48..
    v2f b;
    b.x = bcol[(size_t)(k)     * NCOL];
    b.y = bcol[(size_t)(k + 1) * NCOL];
    // 8 args: (neg_a, A, neg_b, B, c_mod, C, reuse_a, reuse_b)
    acc0 = __builtin_amdgcn_wmma_f32_16x16x4_f32(false, a0, false, b, (short)0, acc0, false, false);
    acc1 = __builtin_amdgcn_wmma_f32_16x16x4_f32(false, a1, false, b, (short)0, acc1, false, false);
    acc2 = __builtin_amdgcn_wmma_f32_16x16x4_f32(false, a2, false, b, (short)0, acc2, false, false);
    acc3 = __builtin_amdgcn_wmma_f32_16x16x4_f32(false, a3, false, b, (short)0, acc3, false, false);
  }

  float* crow = M + (size_t)(m0 + 8 * hi) * NCOL + (n0 + l);
#pragma unroll
  for (int r = 0; r < 8; ++r) crow[(size_t)(r)      * NCOL] = acc0[r];
#pragma unroll
  for (int r = 0; r < 8; ++r) crow[(size_t)(r + 16) * NCOL] = acc1[r];
#pragma unroll
  for (int r = 0; r < 8; ++r) crow[(size_t)(r + 32) * NCOL] = acc2[r];
#pragma unroll
  for (int r = 0; r < 8; ++r) crow[(size_t)(r + 48) * NCOL] = acc3[r];
}

// ---------------------------------------------------------------------------
// Kernel 2: one block per output feature o.
// o_b[i,o] = sum_j exp(-sum_k |M[i,o,k]-M[j,o,k]|) - 1
// 256 threads: thread t handles row i = t&127, j-half = t>>7 (64 j's each).
// LDS: 128x32 f32 slice (16 KB) + 256-float partial buffer.
// ---------------------------------------------------------------------------
__global__ __launch_bounds__(256) void pairwise_l1_exp(
    const float* __restrict__ M, float* __restrict__ out) {
  __shared__ float tile[B_SZ][INTER_F];    // 128B row stride -> b128-friendly
  __shared__ float partial[256];

  const int o   = blockIdx.x;
  const int tid = threadIdx.x;

  // Stage M[:, o, :] into LDS (16 elements per thread, vectorizable).
  for (int idx = tid; idx < B_SZ * INTER_F; idx += 256) {
    const int i  = idx >> 5;
    const int kk = idx & 31;
    tile[i][kk] = M[(size_t)i * NCOL + o * INTER_F + kk];
  }
  __syncthreads();

  const int i    = tid & 127;
  const int hsel = tid >> 7;
  float mi[INTER_F];
#pragma unroll
  for (int kk = 0; kk < INTER_F; ++kk) mi[kk] = tile[i][kk];

  float acc = 0.0f;
  const int jbase = hsel * 64;
  for (int j = jbase; j < jbase + 64; ++j) {
    float s = 0.0f;
#pragma unroll
    for (int kk = 0; kk < INTER_F; ++kk)
      s += fabsf(mi[kk] - tile[j][kk]);    // broadcast LDS read (same j/wave)
    acc += __expf(-s);                     // v_exp_f32 (transcendental)
  }
  partial[tid] = acc;
  __syncthreads();

  if (tid < B_SZ) {
    const float v = partial[tid] + partial[tid + B_SZ] - 1.0f;  // minus self
    out[(size_t)tid * OUTW + IN_F + o] = v;
  }
}

// ---------------------------------------------------------------------------
// Kernel 3: out[:, 0:1024] = x  (one block per row)
// ---------------------------------------------------------------------------
__global__ __launch_bounds__(256) void copy_x(
    const float* __restrict__ x, float* __restrict__ out) {
  const int row = blockIdx.x;
  for (int c = threadIdx.x; c < IN_F; c += 256)
    out[(size_t)row * OUTW + c] = x[(size_t)row * IN_F + c];
}

extern "C" void kernel_launch(void* const* d_in, const int* in_sizes, int n_in,
                              void* d_out, int out_size, void* d_ws, size_t ws_size,
                              hipStream_t stream) {
  (void)in_sizes; (void)n_in; (void)out_size; (void)ws_size;
  const float* x = (const float*)d_in[0];     // [128, 1024]
  const float* T = (const float*)d_in[1];     // [1024, 4096]
  float* out = (float*)d_out;                 // [128, 1152]
  float* M   = (float*)d_ws;                  // scratch: 128*4096*4 = 2 MB

  gemm_wmma_f32  <<<NCOL / 64, 256, 0, stream>>>(x, T, M);
  pairwise_l1_exp<<<OUT_F,     256, 0, stream>>>(M, out);
  copy_x         <<<B_SZ,      256, 0, stream>>>(x, out);
}